// RelationalAttentionLayer_2456721293918
// MI455X (gfx1250) — compile-verified
//
#include <hip/hip_runtime.h>
#include <hip/hip_bf16.h>
#include <math.h>

// ---------------------------------------------------------------------------
// GAT layer, fused single edge-pass on MI455X (gfx1250, wave32, WMMA f32).
//   c = h @ W.T + b                  [E,192]x[192,128] via V_WMMA_F32_16X16X4
//   a = sum(w_att * c) per head ; e = exp(-leaky_relu(a))
//   h_acc[dst] += e * c   (atomics, L2-resident 25.6MB)
//   rowsum[dst][head] += e
//   out = elu(h_acc / rowsum)
// Round 3: double-buffered GLOBAL_LOAD_ASYNC_TO_LDS_B128 prefetch of h tiles
// (ASYNCcnt-tracked), overlapping HBM tile fetch with the WMMA GEMM.
// ---------------------------------------------------------------------------

typedef float v2f __attribute__((ext_vector_type(2)));
typedef float v8f __attribute__((ext_vector_type(8)));
typedef int   v4i __attribute__((ext_vector_type(4)));

#define HEADS     2
#define OUTD      64
#define DIN       192          // 2*64 + 64
#define NOUT      128          // HEADS*OUTD
#define KSTEPS    48           // 192 / 4
#define HSTR      196          // LDS stride (mod 64 == 4 -> conflict free)
#define WSTR      196
#define NEG_SLOPE 0.2f

#if defined(__gfx1250__) && __has_builtin(__builtin_amdgcn_global_load_async_to_lds_b128)
#define HAVE_ASYNC_LDS 1
#else
#define HAVE_ASYNC_LDS 0
#endif

#if HAVE_ASYNC_LDS
typedef __attribute__((address_space(1))) v4i* gas_v4i_ptr;
typedef __attribute__((address_space(3))) v4i* lds_v4i_ptr;

#if __has_builtin(__builtin_amdgcn_s_wait_asynccnt)
#define WAIT_ASYNC(n) __builtin_amdgcn_s_wait_asynccnt(n)
#else
#define WAIT_ASYNC(n) asm volatile("s_wait_asynccnt %0" :: "i"(n) : "memory")
#endif

// One tile = 16 edges x 192 floats = 768 float4 = 3 b128 async loads per thread.
__device__ __forceinline__ void issue_async_tile(const float* __restrict__ gsrc,
                                                 float* __restrict__ dstbuf,
                                                 int tid)
{
    #pragma unroll
    for (int it = 0; it < 3; ++it) {
        int j = tid + it * 256;        // float4 index, 0..767
        int f = j * 4;
        int m = f / DIN, k = f % DIN;
        void* gp = (void*)(gsrc + f);
        void* lp = (void*)(dstbuf + m * HSTR + k);
        __builtin_amdgcn_global_load_async_to_lds_b128(
            (gas_v4i_ptr)gp, (lds_v4i_ptr)lp, 0, 0);
    }
}
#endif // HAVE_ASYNC_LDS

__global__ __launch_bounds__(256, 2)
void gat_edge_pass(const float* __restrict__ h_ijk,
                   const float* __restrict__ W,
                   const float* __restrict__ bias,
                   const float* __restrict__ w_att,
                   const int*   __restrict__ ends,
                   float* __restrict__ h_acc,     // [n,128] numerator accumulator
                   float* __restrict__ rowsum,    // [n,2]
                   int E, int numTiles)
{
    __shared__ float w_lds[NOUT * WSTR];     // 100,352 B : full W, staged once
    __shared__ float h_lds[2][16 * HSTR];    //  25,088 B : double-buffered tiles
    __shared__ float a_lds[16 * HEADS];      // attention logits accumulators
    __shared__ float e_lds[16 * HEADS];      // exp(logits)
    __shared__ int   ends_lds[16];

    const int tid  = threadIdx.x;
    const int lane = tid & 31;
    const int wv   = tid >> 5;          // 8 waves
    const int nb   = wv * 16;           // N-tile base for this wave
    const int nl   = lane & 15;
    const int hi   = lane >> 4;         // half-wave id (M split / K split)
    const int head = nb >> 6;           // 0 for waves 0..3, 1 for waves 4..7
    const int nglob = nb + nl;
    const int stride = gridDim.x;

    int  t  = blockIdx.x;
    int  pb = 0;

#if HAVE_ASYNC_LDS
    // Prologue: kick off first tile's async copy before staging W.
    bool curPref = false;
    if (t < numTiles && (t * 16 + 16) <= E) {
        issue_async_tile(h_ijk + (size_t)t * 16 * DIN, h_lds[0], tid);
        curPref = true;
    }
#endif

    // ---- stage all of W into LDS once (row-major [128][192] -> stride 196) ----
    for (int j = tid; j < (NOUT * DIN) / 4; j += 256) {
        float4 v = ((const float4*)W)[j];
        int f = j * 4;
        int r = f / DIN, c = f % DIN;
        *(float4*)&w_lds[r * WSTR + c] = v;
    }
    const float bval = bias[nglob];
    const float watt = w_att[head * OUTD + (nglob & (OUTD - 1))];
    __syncthreads();

    for (; t < numTiles; t += stride) {
        const int base  = t * 16;
        const int valid = min(16, E - base);
        const int tn    = t + stride;

        __syncthreads();   // prev tile's LDS reads done before buffer reuse

#if HAVE_ASYNC_LDS
        const bool nextPref = (tn < numTiles) && ((tn * 16 + 16) <= E);
        if (nextPref)
            issue_async_tile(h_ijk + (size_t)tn * 16 * DIN, h_lds[1 - pb], tid);
        if (curPref) {
            // 3 outstanding async instrs belong to the prefetched next tile.
            if (nextPref) { WAIT_ASYNC(3); }
            else          { WAIT_ASYNC(0); }
        } else {
            // partial tail tile: synchronous guarded load
            const float* hsrc = h_ijk + (size_t)base * DIN;
            for (int j = tid; j < 16 * DIN; j += 256) {
                int m = j / DIN, k = j % DIN;
                h_lds[pb][m * HSTR + k] = (m < valid) ? hsrc[j] : 0.0f;
            }
        }
#else
        // ---- synchronous load of 16x192 h tile (contiguous 3072 floats) ----
        const float* hsrc = h_ijk + (size_t)base * DIN;
        if (valid == 16) {
            for (int j = tid; j < (16 * DIN) / 4; j += 256) {
                float4 v = ((const float4*)hsrc)[j];
                int f = j * 4;
                int m = f / DIN, k = f % DIN;
                *(float4*)&h_lds[pb][m * HSTR + k] = v;
            }
        } else {
            for (int j = tid; j < 16 * DIN; j += 256) {
                int m = j / DIN, k = j % DIN;
                h_lds[pb][m * HSTR + k] = (m < valid) ? hsrc[j] : 0.0f;
            }
        }
#endif
        if (tid < 16 * HEADS) a_lds[tid] = 0.0f;
        if (tid < 16) ends_lds[tid] = (tid < valid) ? ends[base + tid] : 0;
        __syncthreads();

        // ---- GEMM: D[16x16] = A[16x4] x B[4x16] over 48 K-steps ----
        // A layout (f32 16x4): lanes 0-15 K=0,1 ; lanes 16-31 K=2,3 ; M = lane&15
        // B layout (4x16):     v0 rows K=0/K=2 ; v1 rows K=1/K=3    ; N = lane&15
        v8f acc = {bval, bval, bval, bval, bval, bval, bval, bval};
        const float* arow = &h_lds[pb][nl * HSTR + hi * 2];
        const float* brow = &w_lds[nglob * WSTR + hi * 2];
        #pragma unroll
        for (int kk = 0; kk < KSTEPS; ++kk) {
            v2f af = *(const v2f*)(arow + kk * 4);
            v2f bf = *(const v2f*)(brow + kk * 4);
            acc = __builtin_amdgcn_wmma_f32_16x16x4_f32(
                false, af, false, bf, (short)0, acc, false, false);
        }

        // ---- attention logits: a[m][head] = sum_n w_att[n] * c[m][n] ----
        // C layout: vgpr r -> M = r + 8*hi ; N = lane&15. Butterfly over the
        // 16 lanes of each half (xor 1/2/4/8 never crosses bit 4).
        #pragma unroll
        for (int r = 0; r < 8; ++r) {
            float p = watt * acc[r];
            p += __shfl_xor(p, 1);
            p += __shfl_xor(p, 2);
            p += __shfl_xor(p, 4);
            p += __shfl_xor(p, 8);
            if (nl == 0)
                atomicAdd(&a_lds[(hi * 8 + r) * HEADS + head], p);
        }
        __syncthreads();

        // ---- e = exp(-leaky_relu(a)) ; rowsum scatter (one wave) ----
        if (wv == 0) {
            int m  = lane & 15;
            int hh = lane >> 4;
            float av = a_lds[m * HEADS + hh];
            float lg = -(av >= 0.0f ? av : NEG_SLOPE * av);
            float ev = expf(lg);
            e_lds[m * HEADS + hh] = ev;
            if (m < valid)
                atomicAdd(&rowsum[(size_t)ends_lds[m] * HEADS + hh], ev);
        }
        __syncthreads();

        // ---- numerator scatter: h_acc[dst][n] += e[m] * c[m][n] ----
        // lanes 0-15 of each r hit 16 consecutive floats of one node's row.
        #pragma unroll
        for (int r = 0; r < 8; ++r) {
            int m = hi * 8 + r;
            if (m < valid) {
                float ev = e_lds[m * HEADS + head];
                atomicAdd(&h_acc[(size_t)ends_lds[m] * NOUT + nglob], ev * acc[r]);
            }
        }

#if HAVE_ASYNC_LDS
        curPref = nextPref;
#endif
        pb ^= 1;
    }
}

__global__ void gat_zero(float* __restrict__ ws, long total)
{
    long i = (long)blockIdx.x * blockDim.x + threadIdx.x;
    if (i < total) ws[i] = 0.0f;
}

__global__ void gat_finalize(const float* __restrict__ h_acc,
                             const float* __restrict__ rowsum,
                             float* __restrict__ out, int n)
{
    int i = blockIdx.x * blockDim.x + threadIdx.x;
    int total = n * NOUT;
    if (i >= total) return;
    int node = i >> 7;          // /128
    int col  = i & (NOUT - 1);
    int head = col >> 6;
    float rs = rowsum[node * HEADS + head];
    float v  = (rs > 0.0f) ? (h_acc[i] / rs) : 0.0f;
    out[i] = (v > 0.0f) ? v : expm1f(v);   // elu, alpha=1
}

extern "C" void kernel_launch(void* const* d_in, const int* in_sizes, int n_in,
                              void* d_out, int out_size, void* d_ws, size_t ws_size,
                              hipStream_t stream)
{
    const float* h_ijk = (const float*)d_in[0];
    const float* W     = (const float*)d_in[1];
    const float* bias  = (const float*)d_in[2];
    const float* w_att = (const float*)d_in[3];
    const int*   ends  = (const int*)d_in[4];

    const int E = in_sizes[0] / DIN;        // 1,600,000
    const int n = out_size / NOUT;          // 50,000

    float* h_acc  = (float*)d_ws;                       // n*128 floats
    float* rowsum = h_acc + (size_t)n * NOUT;           // n*2  floats
    float* out    = (float*)d_out;

    // 1) zero accumulators (must happen every call)
    long zcount = (long)n * NOUT + (long)n * HEADS;
    gat_zero<<<(int)((zcount + 255) / 256), 256, 0, stream>>>((float*)d_ws, zcount);

    // 2) fused edge pass (persistent workgroups; W staged in LDS once each;
    //    double-buffered async global->LDS tile prefetch)
    const int numTiles = (E + 15) / 16;
    int blocks = numTiles < 4096 ? numTiles : 4096;
    gat_edge_pass<<<blocks, 256, 0, stream>>>(h_ijk, W, bias, w_att, ends,
                                              h_acc, rowsum, E, numTiles);

    // 3) normalize + elu
    int total = n * NOUT;
    gat_finalize<<<(total + 255) / 256, 256, 0, stream>>>(h_acc, rowsum, out, n);
}